// BackboneModule_89704686944728
// MI455X (gfx1250) — compile-verified
//
#include <hip/hip_runtime.h>
#include <hip/hip_bf16.h>
#include <stdint.h>

#define N_RES_C   300000
#define N_RT      21
#define MAX_RIGID 8
#define MAX_ATOM  24
#define BLOCK     128
// Padded LDS strides (dwords) so random residue-type gathers spread across 64 banks.
#define T_STRIDE  97   // per-type transforms: 96 dwords used (8 x 4x3)
#define G_STRIDE  73   // per-type atom coords: 72 dwords used (24 x 3)

#if defined(__has_builtin)
#if __has_builtin(__builtin_amdgcn_global_load_async_to_lds_b32)
#define HAVE_ASYNC_LDS 1
#endif
#if __has_builtin(__builtin_amdgcn_s_wait_asynccnt)
#define HAVE_WAIT_ASYNC 1
#endif
#endif

typedef __attribute__((address_space(1))) int gint_as1_t;
typedef __attribute__((address_space(3))) int lint_as3_t;

__device__ __forceinline__ void async_cp_dword(const float* gsrc, float* ldst) {
#ifdef HAVE_ASYNC_LDS
  __builtin_amdgcn_global_load_async_to_lds_b32(
      (gint_as1_t*)(int*)(float*)gsrc, (lint_as3_t*)(int*)ldst, 0, 0);
#else
  *ldst = *gsrc;
#endif
}

__device__ __forceinline__ void wait_async_copies() {
#ifdef HAVE_ASYNC_LDS
#ifdef HAVE_WAIT_ASYNC
  __builtin_amdgcn_s_wait_asynccnt(0);
#else
  asm volatile("s_wait_asynccnt 0" ::: "memory");
#endif
#endif
}

__global__ __launch_bounds__(BLOCK) void backbone_kernel(
    const float* __restrict__ bb,     // [N,7]
    const float* __restrict__ sc,     // [N,7,2]
    const float* __restrict__ pos0,   // [N,3]
    const float* __restrict__ Tt,     // [21,8,4,3]
    const float* __restrict__ Gt,     // [21,24,3]
    const int*   __restrict__ rtype,  // [N]
    const int*   __restrict__ tdep,   // [21,8]
    const int*   __restrict__ rdep,   // [21,24]
    float* __restrict__ outR,         // [N,24,3]
    float* __restrict__ outO,         // [N,4,3]
    int nres)
{
  __shared__ float Ts[N_RT * T_STRIDE];
  __shared__ float Gs[N_RT * G_STRIDE];
  __shared__ unsigned char tds[N_RT * MAX_RIGID];
  __shared__ unsigned char rds[N_RT * MAX_ATOM];
  // Per-thread 8x(4x3) frames, transposed layout -> conflict-free DS access.
  __shared__ float oprS[96 * BLOCK];

  const int tid = threadIdx.x;

  // ---- stage per-residue-type tables into LDS (async on CDNA5) ----
  for (int i = tid; i < N_RT * 96; i += BLOCK) {
    int rt = i / 96;
    async_cp_dword(&Tt[i], &Ts[rt * T_STRIDE + (i - rt * 96)]);
  }
  for (int i = tid; i < N_RT * 72; i += BLOCK) {
    int rt = i / 72;
    async_cp_dword(&Gt[i], &Gs[rt * G_STRIDE + (i - rt * 72)]);
  }
  for (int i = tid; i < N_RT * MAX_RIGID; i += BLOCK) tds[i] = (unsigned char)tdep[i];
  for (int i = tid; i < N_RT * MAX_ATOM;  i += BLOCK) rds[i] = (unsigned char)rdep[i];
  wait_async_copies();
  __syncthreads();

  const int r = blockIdx.x * BLOCK + tid;
  if (r >= nres) return;

  const int rt = rtype[r];
  const float* bp = bb   + (size_t)r * 7;
  const float* pp = pos0 + (size_t)r * 3;
  const float* sp = sc   + (size_t)r * 14;

  // Pull the next tile's streaming inputs toward L2 (global_prefetch_b8).
  if (r + BLOCK < nres) {
    __builtin_prefetch(bb + (size_t)(r + BLOCK) * 7, 0, 1);
    __builtin_prefetch(sc + (size_t)(r + BLOCK) * 14, 0, 1);
  }

  float opr[MAX_RIGID][12];   // rows 0..2 = R, row 3 (idx 9..11) = t

  // ---- backbone frame: raw-quat -> rot (reference does not normalize) ----
  {
    float w = bp[0], x = bp[1], y = bp[2], z = bp[3];
    float ww = w*w, xx = x*x, yy = y*y, zz = z*z;
    float xy = x*y, wz = w*z, xz = x*z, wy = w*y, yz = y*z, wx = w*x;
    float L[12];
    L[0] = ww + xx - yy - zz;  L[1] = 2.f*(xy - wz);      L[2] = 2.f*(xz + wy);
    L[3] = 2.f*(xy + wz);      L[4] = ww - xx + yy - zz;  L[5] = 2.f*(yz - wx);
    L[6] = 2.f*(xz - wy);      L[7] = 2.f*(yz + wx);      L[8] = ww - xx - yy + zz;
    L[9]  = 0.1f*bp[4] + pp[0];
    L[10] = 0.1f*bp[5] + pp[1];
    L[11] = 0.1f*bp[6] + pp[2];
    const float* X = &Ts[rt * T_STRIDE];          // type transform 0
    #pragma unroll
    for (int a = 0; a < 3; ++a) {                 // opr[0] = combine(X, L)
      float xa0 = X[a*3+0], xa1 = X[a*3+1], xa2 = X[a*3+2];
      #pragma unroll
      for (int b = 0; b < 3; ++b)
        opr[0][a*3+b] = xa0*L[b] + xa1*L[3+b] + xa2*L[6+b];
      opr[0][9+a] = xa0*L[9] + xa1*L[10] + xa2*L[11] + X[9+a];
    }
  }

  // ---- torsion frames pre-combined with table transforms (Y_t == 0) ----
  #pragma unroll
  for (int k = 1; k < MAX_RIGID; ++k) {
    float c = sp[2*(k-1)], s = sp[2*(k-1)+1];
    const float* X = &Ts[rt * T_STRIDE + k*12];
    #pragma unroll
    for (int a = 0; a < 3; ++a) {
      float xa1 = X[a*3+1], xa2 = X[a*3+2];
      opr[k][a*3+0] = X[a*3+0];
      opr[k][a*3+1] = xa1*c + xa2*s;
      opr[k][a*3+2] = xa2*c - xa1*s;
      opr[k][9+a]   = X[9+a];
    }
  }

  // ---- dependency chain: opr[i] = combine(opr[tdep[i]], opr[i]), tdep[i] < i ----
  #pragma unroll
  for (int i = 1; i < MAX_RIGID; ++i) {
    int td = tds[rt * MAX_RIGID + i];
    float p[12];
    #pragma unroll
    for (int d = 0; d < 12; ++d) p[d] = opr[0][d];
    #pragma unroll
    for (int j = 1; j < i; ++j) {
      bool sel = (td == j);
      #pragma unroll
      for (int d = 0; d < 12; ++d) p[d] = sel ? opr[j][d] : p[d];
    }
    float y[12];
    #pragma unroll
    for (int d = 0; d < 12; ++d) y[d] = opr[i][d];
    #pragma unroll
    for (int a = 0; a < 3; ++a) {
      float pa0 = p[a*3+0], pa1 = p[a*3+1], pa2 = p[a*3+2];
      #pragma unroll
      for (int b = 0; b < 3; ++b)
        opr[i][a*3+b] = pa0*y[b] + pa1*y[3+b] + pa2*y[6+b];
      opr[i][9+a] = pa0*y[9] + pa1*y[10] + pa2*y[11] + p[9+a];
    }
  }

  // ---- second output: combined backbone frame, 16B vector stores ----
  {
    float4* o = (float4*)(outO + (size_t)r * 12);
    o[0] = make_float4(opr[0][0], opr[0][1], opr[0][2],  opr[0][3]);
    o[1] = make_float4(opr[0][4], opr[0][5], opr[0][6],  opr[0][7]);
    o[2] = make_float4(opr[0][8], opr[0][9], opr[0][10], opr[0][11]);
  }

  // ---- park frames in LDS for the data-dependent per-atom gather ----
  #pragma unroll
  for (int k = 0; k < MAX_RIGID; ++k) {
    #pragma unroll
    for (int d = 0; d < 12; ++d)
      oprS[(k*12 + d) * BLOCK + tid] = opr[k][d];
  }

  // ---- atom placement: R[a] = g_R @ G[a] + g_t, g = opr[rdep[a]] ----
  // Accumulate all 72 output floats in registers, then emit 18 aligned b128 stores.
  float outbuf[MAX_ATOM * 3];
  #pragma unroll
  for (int a = 0; a < MAX_ATOM; ++a) {
    int k = rds[rt * MAX_ATOM + a];
    int base = k * 12 * BLOCK + tid;   // k-stride 12*128 ≡ 0 mod 64 -> conflict-free
    float g[12];
    #pragma unroll
    for (int d = 0; d < 12; ++d) g[d] = oprS[base + d * BLOCK];
    float Gx = Gs[rt * G_STRIDE + a*3 + 0];
    float Gy = Gs[rt * G_STRIDE + a*3 + 1];
    float Gz = Gs[rt * G_STRIDE + a*3 + 2];
    outbuf[a*3+0] = g[0]*Gx + g[1]*Gy + g[2]*Gz + g[9];
    outbuf[a*3+1] = g[3]*Gx + g[4]*Gy + g[5]*Gz + g[10];
    outbuf[a*3+2] = g[6]*Gx + g[7]*Gy + g[8]*Gz + g[11];
  }
  {
    float4* o = (float4*)(outR + (size_t)r * (MAX_ATOM * 3));  // r*288B, 16B aligned
    #pragma unroll
    for (int v = 0; v < (MAX_ATOM * 3) / 4; ++v)
      o[v] = make_float4(outbuf[4*v+0], outbuf[4*v+1], outbuf[4*v+2], outbuf[4*v+3]);
  }
}

extern "C" void kernel_launch(void* const* d_in, const int* in_sizes, int n_in,
                              void* d_out, int out_size, void* d_ws, size_t ws_size,
                              hipStream_t stream) {
  const float* bb    = (const float*)d_in[0];
  const float* sc    = (const float*)d_in[1];
  const float* pos0  = (const float*)d_in[2];
  const float* Tt    = (const float*)d_in[3];
  const float* Gt    = (const float*)d_in[4];
  const int*   rtype = (const int*)d_in[5];
  const int*   tdep  = (const int*)d_in[6];
  const int*   rdep  = (const int*)d_in[7];

  const int nres = in_sizes[5];                  // residue_type has N_RES elements
  float* outR = (float*)d_out;                   // [N,24,3] first
  float* outO = outR + (size_t)nres * MAX_ATOM * 3;  // then [N,4,3]

  int grid = (nres + BLOCK - 1) / BLOCK;
  backbone_kernel<<<dim3(grid), dim3(BLOCK), 0, stream>>>(
      bb, sc, pos0, Tt, Gt, rtype, tdep, rdep, outR, outO, nres);
}